// DepthConv_1786706395749
// MI455X (gfx1250) — compile-verified
//
#include <hip/hip_runtime.h>

// Involution-style per-pixel depthwise 3x3 conv (pad=1, stride=1), fp32.
//   out[n,c,h,w] = sum_{k=0..8} wts[n,c,k,h,w] * x_pad[n,c,h+k/3, w+k%3]
//
// Memory-bound (~0.41 FLOP/byte): weights dominate (9 planes, each used once).
// Strategy: one block per (n,c) plane; stage the 36KB x plane into LDS with
// gfx1250 async global->LDS DMA; stream weights with non-temporal b128 loads;
// keep the 3-row x window in registers for vertical reuse.

namespace {

constexpr int HH   = 96;
constexpr int WW   = 96;
constexpr int HWP  = HH * WW;            // 9216 elements per plane
constexpr int NTAPS = 9;

constexpr int BLOCKT = 192;              // 6 waves of 32
constexpr int STRIPS = WW / 4;           // 24 col strips of width 4
constexpr int BAND_H = HH / (BLOCKT / STRIPS); // 12 rows per thread
constexpr int CHUNKS = HWP / 4;          // 2304 16-byte chunks per plane
constexpr int CHUNKS_PER_THR = CHUNKS / BLOCKT; // 12

typedef float float4v __attribute__((ext_vector_type(4)));
typedef int   int4v   __attribute__((ext_vector_type(4)));

} // namespace

__global__ __launch_bounds__(BLOCKT) void involution_dw3x3_kernel(
    const float* __restrict__ x, const float* __restrict__ wts,
    float* __restrict__ out) {
  __shared__ float xs[HWP];              // 36 KB of the 320 KB WGP LDS

  const int plane = (int)blockIdx.x;     // n*C + c
  const int tid   = (int)threadIdx.x;

  const float* xp = x   + (size_t)plane * HWP;
  const float* wp = wts + (size_t)plane * (size_t)(NTAPS * HWP);
  float*       op = out + (size_t)plane * HWP;

  // ---- Stage this plane of x into LDS ----------------------------------
#if defined(__AMDGCN__) &&                                           \
    __has_builtin(__builtin_amdgcn_global_load_async_to_lds_b128) && \
    __has_builtin(__builtin_amdgcn_s_wait_asynccnt)
  {
    // gfx1250 async DMA path: per-lane 16B transfers, tracked by ASYNCcnt,
    // bypasses VGPRs entirely. Builtin expects AS1 / AS3 pointers to a
    // 16-byte int vector.
    typedef __attribute__((address_space(1))) int4v gint4;
    typedef __attribute__((address_space(3))) int4v lint4;
    gint4* gsrc = (gint4*)xp;
    lint4* ldst = (lint4*)xs;
#pragma unroll
    for (int i = 0; i < CHUNKS_PER_THR; ++i) {
      const int chunk = tid + i * BLOCKT;
      __builtin_amdgcn_global_load_async_to_lds_b128(gsrc + chunk,
                                                     ldst + chunk,
                                                     /*offset=*/0,
                                                     /*cpol=*/0);
    }
    __builtin_amdgcn_s_wait_asynccnt(0);   // my transfers done
  }
#else
  {
#pragma unroll
    for (int i = 0; i < CHUNKS_PER_THR; ++i) {
      const int chunk = tid + i * BLOCKT;
      ((float4v*)xs)[chunk] = ((const float4v*)xp)[chunk];
    }
  }
#endif
  __syncthreads();                        // everyone's transfers done

  // ---- Thread -> (row band, col strip) ---------------------------------
  const int strip = tid % STRIPS;         // 0..23
  const int band  = tid / STRIPS;         // 0..7
  const int w0    = strip * 4;            // first output column (mult of 4)
  const int r0    = band * BAND_H;        // first output row

  // Sliding 3-row x window, columns w0-1 .. w0+4 (zero at pad).
  float rm1[6], rc[6], rp1[6];

  auto loadrow = [&](int r, float* dst) {
    if ((unsigned)r < (unsigned)HH) {
      const float* row = &xs[r * WW];
#pragma unroll
      for (int c = 0; c < 6; ++c) {
        const int cc = w0 + c - 1;
        dst[c] = ((unsigned)cc < (unsigned)WW) ? row[cc] : 0.0f;
      }
    } else {
#pragma unroll
      for (int c = 0; c < 6; ++c) dst[c] = 0.0f;
    }
  };

  loadrow(r0 - 1, rm1);
  loadrow(r0,     rc);

#pragma unroll
  for (int rr = 0; rr < BAND_H; ++rr) {
    const int r = r0 + rr;
    loadrow(r + 1, rp1);                  // only new row hits LDS (6 reads)
    const int p = r * WW + w0;            // flat pixel offset, 16B aligned

    float4v acc = {0.0f, 0.0f, 0.0f, 0.0f};

    // Tap k = di*3+dj ; output col j reads x window index j+dj.
#define ACC_TAP(K, XR, DJ)                                           \
  do {                                                               \
    float4v wv = __builtin_nontemporal_load(                         \
        (const float4v*)(wp + (size_t)(K) * HWP + p));               \
    acc[0] += wv[0] * (XR)[0 + (DJ)];                                \
    acc[1] += wv[1] * (XR)[1 + (DJ)];                                \
    acc[2] += wv[2] * (XR)[2 + (DJ)];                                \
    acc[3] += wv[3] * (XR)[3 + (DJ)];                                \
  } while (0)

    ACC_TAP(0, rm1, 0);
    ACC_TAP(1, rm1, 1);
    ACC_TAP(2, rm1, 2);
    ACC_TAP(3, rc,  0);
    ACC_TAP(4, rc,  1);
    ACC_TAP(5, rc,  2);
    ACC_TAP(6, rp1, 0);
    ACC_TAP(7, rp1, 1);
    ACC_TAP(8, rp1, 2);
#undef ACC_TAP

    __builtin_nontemporal_store(acc, (float4v*)(op + p));

    // Slide window down one row (register renaming after unroll).
#pragma unroll
    for (int c = 0; c < 6; ++c) {
      rm1[c] = rc[c];
      rc[c]  = rp1[c];
    }
  }
}

extern "C" void kernel_launch(void* const* d_in, const int* in_sizes, int n_in,
                              void* d_out, int out_size, void* d_ws,
                              size_t ws_size, hipStream_t stream) {
  (void)in_sizes; (void)n_in; (void)d_ws; (void)ws_size;
  const float* x   = (const float*)d_in[0];   // (N, C, H, W) fp32
  const float* wts = (const float*)d_in[1];   // (N, C*9, H, W) fp32
  float* out       = (float*)d_out;           // (N, C, H, W) fp32

  const int planes = out_size / HWP;          // N*C = 1024
  involution_dw3x3_kernel<<<dim3(planes), dim3(BLOCKT), 0, stream>>>(x, wts,
                                                                     out);
}